// ClusterLayer_58514634440868
// MI455X (gfx1250) — compile-verified
//
#include <hip/hip_runtime.h>

#define N_ 16384
#define K_ 64
#define F_ 2048
#define ALPHA_ 1.0f
#define GAMMA_ 1.0f

typedef float v2f __attribute__((ext_vector_type(2)));
typedef float v8f __attribute__((ext_vector_type(8)));

// ---------------------------------------------------------------------------
// Kernel 1: G = C * C^T  (64x64, K-dim = 2048) via V_WMMA_F32_16X16X4_F32.
// One wave (32 threads) per 16x16 output tile; 16 tiles total.
// f32 16x4 A layout: lanes 0-15 hold M=0..15 with VGPR{0,1}=K{0,1};
// lanes 16-31 hold K{2,3}  ->  per-lane b64 load of 2 consecutive floats.
// B (4x16) mirrors this since both tiles are rows of row-major C.
// ---------------------------------------------------------------------------
__global__ __launch_bounds__(32) void gram_wmma(const float* __restrict__ C,
                                                float* __restrict__ G) {
    const int tile = blockIdx.x;   // 0..15
    const int ta = tile >> 2;      // row tile of G
    const int tb = tile & 3;       // col tile of G
    const int lane = threadIdx.x;  // 0..31 (wave32)
    const int lo = lane & 15;
    const int hi = lane >> 4;

    const v2f* __restrict__ ap = (const v2f*)(C + (size_t)(ta * 16 + lo) * F_);
    const v2f* __restrict__ bp = (const v2f*)(C + (size_t)(tb * 16 + lo) * F_);

    v8f acc = {0.f, 0.f, 0.f, 0.f, 0.f, 0.f, 0.f, 0.f};
#pragma unroll 4
    for (int i = 0; i < F_ / 4; ++i) {      // 512 WMMA accumulations
        v2f a = ap[2 * i + hi];             // K = 4i + 2*hi + {0,1}
        v2f b = bp[2 * i + hi];
        acc = __builtin_amdgcn_wmma_f32_16x16x4_f32(
            /*neg_a=*/false, a, /*neg_b=*/false, b,
            /*c_mod=*/(short)0, acc, /*reuse_a=*/false, /*reuse_b=*/false);
    }
    // C/D layout: lanes 0-15: VGPR r -> M=r; lanes 16-31: VGPR r -> M=8+r; N=lane%16
#pragma unroll
    for (int r = 0; r < 8; ++r) {
        int m = r + 8 * hi;
        G[(ta * 16 + m) * K_ + tb * 16 + lo] = acc[r];
    }
}

// ---------------------------------------------------------------------------
// Kernel 2: row_sum[k] = K*G[k,k] + sum_j G[j,j] - 2*sum_j G[k,j]
//           rs1[k] = row_sum[k] + alpha ; w[k] = class_weight[k] / rs1[k]
// ---------------------------------------------------------------------------
__global__ __launch_bounds__(64) void stats(const float* __restrict__ G,
                                            const float* __restrict__ cw,
                                            float* __restrict__ rs1,
                                            float* __restrict__ w) {
    __shared__ float nrm_s[K_];
    const int k = threadIdx.x;
    float rowG = 0.f;
#pragma unroll 8
    for (int j = 0; j < K_; ++j) rowG += G[k * K_ + j];
    const float nrm = G[k * K_ + k];
    nrm_s[k] = nrm;
    __syncthreads();
    float S = 0.f;
#pragma unroll 8
    for (int j = 0; j < K_; ++j) S += nrm_s[j];
    const float rs = (float)K_ * nrm + S - 2.f * rowG + ALPHA_;
    rs1[k] = rs;
    w[k] = cw[k] / rs;
}

// ---------------------------------------------------------------------------
// Kernel 3: per-row argmax(labels) -> idx, then loss[n] = ||x-c_idx||^2 / rs1.
// One wave32 per row; 8 rows per 256-thread block. float4 streaming loads.
// ---------------------------------------------------------------------------
__global__ __launch_bounds__(256) void loss_idx(const float* __restrict__ features,
                                                const float* __restrict__ labels,
                                                const float* __restrict__ cluster,
                                                const float* __restrict__ rs1,
                                                float* __restrict__ loss,
                                                int* __restrict__ idx_out) {
    const int lane = threadIdx.x & 31;
    const int row = blockIdx.x * 8 + (threadIdx.x >> 5);

    // argmax over 64 one-hot labels (first-max semantics like jnp.argmax)
    float v0 = labels[(size_t)row * K_ + lane];
    float v1 = labels[(size_t)row * K_ + lane + 32];
    float bv = v0;
    int bi = lane;
    if (v1 > v0) { bv = v1; bi = lane + 32; }
#pragma unroll
    for (int m = 16; m >= 1; m >>= 1) {
        float ov = __shfl_xor(bv, m, 32);
        int oi = __shfl_xor(bi, m, 32);
        if (ov > bv || (ov == bv && oi < bi)) { bv = ov; bi = oi; }
    }
    const int idx = bi;  // all lanes agree after butterfly

    const float4* __restrict__ x4 = (const float4*)(features + (size_t)row * F_);
    const float4* __restrict__ c4 = (const float4*)(cluster + (size_t)idx * F_);
    float s = 0.f;
#pragma unroll 4
    for (int i = lane; i < F_ / 4; i += 32) {
        float4 a = x4[i];
        float4 b = c4[i];
        float d0 = a.x - b.x, d1 = a.y - b.y, d2 = a.z - b.z, d3 = a.w - b.w;
        s += d0 * d0 + d1 * d1 + d2 * d2 + d3 * d3;
    }
#pragma unroll
    for (int m = 16; m >= 1; m >>= 1) s += __shfl_xor(s, m, 32);

    if (lane == 0) {
        loss[row] = s / rs1[idx];
        idx_out[row] = idx;
    }
}

// ---------------------------------------------------------------------------
// Kernel 4: block (blockIdx.y = k, blockIdx.x = 512-col tile) scans idx[],
// accumulates matching feature rows in registers (no atomics), writes
// new_cluster[k,f] = c - gamma*w[k]*(fsum - count*c).
// ---------------------------------------------------------------------------
__global__ __launch_bounds__(256) void cluster_update(const float* __restrict__ features,
                                                      const float* __restrict__ cluster,
                                                      const float* __restrict__ w,
                                                      const int* __restrict__ idx,
                                                      float* __restrict__ newc) {
    const int k = blockIdx.y;
    const int c0 = blockIdx.x * 512 + threadIdx.x * 2;  // 2 columns per thread
    float a0 = 0.f, a1 = 0.f;
    int cnt = 0;
    for (int n = 0; n < N_; ++n) {
        if (idx[n] == k) {  // uniform (scalar) branch across block
            const float2 xr = *(const float2*)(features + (size_t)n * F_ + c0);
            a0 += xr.x;
            a1 += xr.y;
            ++cnt;
        }
    }
    const float wk = w[k];
    const float2 cc = *(const float2*)(cluster + (size_t)k * F_ + c0);
    float2 out;
    out.x = cc.x - GAMMA_ * wk * (a0 - (float)cnt * cc.x);
    out.y = cc.y - GAMMA_ * wk * (a1 - (float)cnt * cc.y);
    *(float2*)(newc + (size_t)k * F_ + c0) = out;
}

// ---------------------------------------------------------------------------
extern "C" void kernel_launch(void* const* d_in, const int* in_sizes, int n_in,
                              void* d_out, int out_size, void* d_ws, size_t ws_size,
                              hipStream_t stream) {
    const float* features = (const float*)d_in[0];  // [N, F]
    const float* labels   = (const float*)d_in[1];  // [N, K] one-hot
    const float* cluster  = (const float*)d_in[2];  // [K, F]
    const float* cw       = (const float*)d_in[3];  // [K]

    float* loss = (float*)d_out;          // [N]
    float* newc = loss + N_;              // [K, F]

    // workspace layout (~82 KB)
    float* G   = (float*)d_ws;            // [K, K] Gram
    float* rs1 = G + K_ * K_;             // [K] row_sum + alpha
    float* wv  = rs1 + K_;                // [K] class_weight / rs1
    int* idxw  = (int*)(wv + K_);         // [N] argmax indices

    gram_wmma<<<16, 32, 0, stream>>>(cluster, G);
    stats<<<1, 64, 0, stream>>>(G, cw, rs1, wv);
    loss_idx<<<N_ / 8, 256, 0, stream>>>(features, labels, cluster, rs1, loss, idxw);
    cluster_update<<<dim3(F_ / 512, K_), 256, 0, stream>>>(features, cluster, wv, idxw, newc);
}